// MaskBfEstimator_14224931684412
// MI455X (gfx1250) — compile-verified
//
#include <hip/hip_runtime.h>
#include <math.h>

#define FRAMES 4096
#define MICS   8
#define NBINS  513

typedef float v2f  __attribute__((ext_vector_type(2)));
typedef float v8f  __attribute__((ext_vector_type(8)));
typedef unsigned int u32x4 __attribute__((ext_vector_type(4)));
typedef int   i32x4 __attribute__((ext_vector_type(4)));
typedef int   i32x8 __attribute__((ext_vector_type(8)));

constexpr int BPB  = 16;              // bins per block, one wave32 per bin
constexpr int CF   = 64;              // frames staged per LDS chunk
constexpr int THREADS_COV = BPB * 32; // 512 threads = 16 waves

// LDS tile geometry with TDM padding: +1 dword after every 16 dwords
// -> mic row stride 17 dwords, breaks LDS bank conflicts (2-way instead of 8-way)
constexpr int MROW    = 17;           // dwords per 16-bin mic row
constexpr int FROW    = MICS * MROW;  // 136 dwords per frame
constexpr int TILE_DW = CF * FROW;    // 8704 dwords per (re/im) tile

#ifdef __has_builtin
#  if __has_builtin(__builtin_amdgcn_tensor_load_to_lds)
#    define HAVE_TDM 1
#  else
#    define HAVE_TDM 0
#  endif
#else
#  define HAVE_TDM 0
#endif
#if __has_include(<hip/amd_detail/amd_gfx1250_TDM.h>)
#  define TDM_6ARG 1   // therock-10.0 headers -> clang-23 6-arg builtin
#else
#  define TDM_6ARG 0   // ROCm 7.2 clang-22 5-arg builtin
#endif

__device__ __forceinline__ float sigmoidf(float v) { return 1.0f / (1.0f + __expf(-v)); }

__device__ __forceinline__ unsigned lds_off(const void* p) {
    // generic pointers to LDS carry the LDS byte offset in the low 32 bits
    return (unsigned)(unsigned long long)p;
}

#if HAVE_TDM
// ---------------------------------------------------------------------------
// Issue one TDM 3-D tile load: (tile_dim0=16 bins, tile_dim1=8 mics,
// tile_dim2=CF frames) from x(F,M,B) into padded LDS. tensor_dim* carry the
// remaining extents from the tile start so the TDM zero-fills OOB bins.
// ---------------------------------------------------------------------------
__device__ __forceinline__ void tdm_load_tile(const float* gbase, int f0, int bin0,
                                              unsigned lds_byte)
{
    unsigned long long ga = (unsigned long long)(size_t)gbase
                          + ((unsigned long long)f0 * (MICS * NBINS) + (unsigned)bin0) * 4ull;
    u32x4 g0;
    g0[0] = 1u;                                                  // count=1, user D#
    g0[1] = lds_byte;                                            // lds_addr
    g0[2] = (unsigned)(ga & 0xffffffffull);                      // global_addr[31:0]
    g0[3] = (unsigned)((ga >> 32) & 0x1ffffffull) | (2u << 30);  // ga[56:32] | type=2

    const unsigned td0 = (unsigned)(NBINS - bin0);   // remaining bins (zero-fill tail)
    const unsigned td1 = MICS;
    const unsigned s0  = NBINS;                      // bin-line stride (elements)
    const unsigned s1  = MICS * NBINS;               // frame stride (elements)
    i32x8 g1;
    g1[0] = (int)((2u << 16)     // data_size = 4 bytes
                | (1u << 20)     // pad_enable
                | (3u << 22));   // pad_interval: every 16 dwords, pad_amount=1 dword
    g1[1] = (int)((td0 & 0xffffu) << 16);                               // dim0[15:0]
    g1[2] = (int)(((td0 >> 16) & 0xffffu) | ((td1 & 0xffffu) << 16));   // dim0 hi|dim1 lo
    g1[3] = (int)(((td1 >> 16) & 0xffffu) | ((unsigned)BPB << 16));     // dim1 hi|tile0
    g1[4] = (int)((unsigned)MICS | ((unsigned)CF << 16));               // tile1 | tile2
    g1[5] = (int)s0;                                                    // stride0 lo
    g1[6] = (int)((s1 & 0xffffu) << 16);                                // stride0 hi|s1 lo
    g1[7] = (int)(s1 >> 16);                                            // stride1 hi

    i32x4 g2;
    g2[0] = FRAMES - f0;   // tensor_dim2 (remaining frames)
    g2[1] = 0;             // tensor_dim3 unused
    g2[2] = 0;             // tensor_dim2_stride unused (tile_dim3 = 0)
    g2[3] = 0;             // tile_dim3 = 0
    i32x4 g3 = {0, 0, 0, 0};

#if TDM_6ARG
    i32x8 g4 = {};
    __builtin_amdgcn_tensor_load_to_lds(g0, g1, g2, g3, g4, 0);
#else
    __builtin_amdgcn_tensor_load_to_lds(g0, g1, g2, g3, 0);
#endif
}
#endif // HAVE_TDM

// Fallback staging (and host-pass parse path): plain coalesced loads into the
// same padded LDS layout.
__device__ __forceinline__ void stage_x_fallback(float* dst, const float* xr,
                                                 const float* xi, int f0, int bin0, int tid)
{
    for (int e = tid; e < CF * MICS * BPB * 2; e += THREADS_COV) {
        int bl = e & 15;
        int m  = (e >> 4) & 7;
        int f  = (e >> 7) & (CF - 1);
        int ri = e >> 13;
        int b  = bin0 + bl;
        const float* src = ri ? xi : xr;
        float v = (b < NBINS) ? src[(size_t)(f0 + f) * (MICS * NBINS) + m * NBINS + b] : 0.0f;
        dst[ri * TILE_DW + f * FROW + m * MROW + bl] = v;
    }
}

// ---------------------------------------------------------------------------
// Kernel 1: masked spatial covariances via V_WMMA_F32_16X16X4_F32, TDM
// double-buffered LDS staging. A=[m*x_re; m*x_im] (16xK), B=[x_re; x_im] (Kx16).
// ---------------------------------------------------------------------------
__global__ __launch_bounds__(THREADS_COV)
void cov_wmma_kernel(const float* __restrict__ xr, const float* __restrict__ xi,
                     const float* __restrict__ masks, float* __restrict__ tiles)
{
    __shared__ float lx[2][2 * TILE_DW];     // [buf][re/im tile, padded]
    __shared__ float lm[CF][2][BPB];         // sigmoid(masks)

    const int tid  = threadIdx.x;
    const int wave = tid >> 5;
    const int lane = tid & 31;
    const int half = lane >> 4;      // K-pair selector per f32 A layout
    const int idx  = lane & 15;      // M (and N): reim*8 + mic
    const int mic  = idx & 7;
    const int reim = idx >> 3;
    const int bin0 = blockIdx.x * BPB;
    const int bin  = bin0 + wave;
    const bool issuer = (wave == 0);

    v8f acc1 = {}, acc2 = {};

#if HAVE_TDM
    if (issuer) {
        tdm_load_tile(xr, 0, bin0, lds_off(&lx[0][0]));
        tdm_load_tile(xi, 0, bin0, lds_off(&lx[0][TILE_DW]));
    }
#endif

    int buf = 0;
    for (int f0 = 0; f0 < FRAMES; f0 += CF, buf ^= 1) {
        for (int e = tid; e < CF * 2 * BPB; e += THREADS_COV) {
            int bl = e & 15, ch = (e >> 4) & 1, f = e >> 5;
            int b = bin0 + bl;
            float v = 0.0f;
            if (b < NBINS) v = sigmoidf(masks[(size_t)(f0 + f) * (2 * NBINS) + ch * NBINS + b]);
            lm[f][ch][bl] = v;
        }
#if HAVE_TDM
        if (issuer) {
            if (f0 + CF < FRAMES) {       // prefetch next chunk, wait only for current
                tdm_load_tile(xr, f0 + CF, bin0, lds_off(&lx[buf ^ 1][0]));
                tdm_load_tile(xi, f0 + CF, bin0, lds_off(&lx[buf ^ 1][TILE_DW]));
                __builtin_amdgcn_s_wait_tensorcnt((short)2);
            } else {
                __builtin_amdgcn_s_wait_tensorcnt((short)0);
            }
        }
#else
        stage_x_fallback(&lx[buf][0], xr, xi, f0, bin0, tid);
#endif
        __syncthreads();

        const float* xs = &lx[buf][reim * TILE_DW];
        #pragma unroll 4
        for (int s = 0; s < CF; s += 4) {
            int k0 = s + half * 2;
            float b0  = xs[(k0)     * FROW + mic * MROW + wave];
            float b1  = xs[(k0 + 1) * FROW + mic * MROW + wave];
            float m10 = lm[k0][0][wave], m11 = lm[k0 + 1][0][wave];
            float m20 = lm[k0][1][wave], m21 = lm[k0 + 1][1][wave];
            v2f B  = {b0, b1};
            v2f A1 = {m10 * b0, m11 * b1};
            v2f A2 = {m20 * b0, m21 * b1};
            acc1 = __builtin_amdgcn_wmma_f32_16x16x4_f32(false, A1, false, B,
                                                         (short)0, acc1, false, false);
            acc2 = __builtin_amdgcn_wmma_f32_16x16x4_f32(false, A2, false, B,
                                                         (short)0, acc2, false, false);
        }
        __syncthreads();
    }

    if (bin < NBINS) {
        const float inv = 1.0f / (float)FRAMES;
        float* t1 = tiles + ((size_t)bin * 2    ) * 256;
        float* t2 = tiles + ((size_t)bin * 2 + 1) * 256;
        #pragma unroll
        for (int r = 0; r < 8; ++r) {
            int M = r + half * 8;
            int N = idx;
            t1[M * 16 + N] = acc1[r] * inv;
            t2[M * 16 + N] = acc2[r] * inv;
        }
    }
}

// ---------------------------------------------------------------------------
// Kernel 2: per-bin 8x8 complex eigen (power iteration, PSD) + Gaussian solve.
// ---------------------------------------------------------------------------
__global__ void solve_kernel(const float* __restrict__ tiles, float* __restrict__ wvec)
{
    int bin = blockIdx.x * blockDim.x + threadIdx.x;
    if (bin >= NBINS) return;
    const float* t1 = tiles + ((size_t)bin * 2) * 256;
    const float* t2 = t1 + 256;

    float c1r[8][8], c1i[8][8], c2r[8][8], c2i[8][8];
    for (int m = 0; m < 8; ++m)
        for (int n = 0; n < 8; ++n) {
            c1r[m][n] = t1[m * 16 + n] + t1[(m + 8) * 16 + (n + 8)];
            c1i[m][n] = t1[(m + 8) * 16 + n] - t1[m * 16 + (n + 8)];
            c2r[m][n] = t2[m * 16 + n] + t2[(m + 8) * 16 + (n + 8)];
            c2i[m][n] = t2[(m + 8) * 16 + n] - t2[m * 16 + (n + 8)];
        }

    float vr[8], vi[8];
    for (int m = 0; m < 8; ++m) { vr[m] = 1.0f; vi[m] = 0.0f; }
    for (int it = 0; it < 32; ++it) {
        float ur[8], ui[8], nrm = 0.0f;
        for (int m = 0; m < 8; ++m) {
            float ar = 0.0f, ai = 0.0f;
            for (int n = 0; n < 8; ++n) {
                ar += c2r[m][n] * vr[n] - c2i[m][n] * vi[n];
                ai += c2r[m][n] * vi[n] + c2i[m][n] * vr[n];
            }
            ur[m] = ar; ui[m] = ai; nrm += ar * ar + ai * ai;
        }
        float rn = rsqrtf(nrm + 1e-30f);
        for (int m = 0; m < 8; ++m) { vr[m] = ur[m] * rn; vi[m] = ui[m] * rn; }
    }

    float ar[8][9], ai[8][9];
    for (int m = 0; m < 8; ++m) {
        for (int n = 0; n < 8; ++n) { ar[m][n] = c1r[m][n]; ai[m][n] = c1i[m][n]; }
        ar[m][8] = vr[m]; ai[m][8] = vi[m];
    }
    for (int k = 0; k < 8; ++k) {
        int p = k;
        float best = ar[k][k] * ar[k][k] + ai[k][k] * ai[k][k];
        for (int r = k + 1; r < 8; ++r) {
            float mag = ar[r][k] * ar[r][k] + ai[r][k] * ai[r][k];
            if (mag > best) { best = mag; p = r; }
        }
        if (p != k)
            for (int c = k; c < 9; ++c) {
                float tr = ar[k][c]; ar[k][c] = ar[p][c]; ar[p][c] = tr;
                float ti = ai[k][c]; ai[k][c] = ai[p][c]; ai[p][c] = ti;
            }
        float dr = ar[k][k], di = ai[k][k];
        float inv = 1.0f / (dr * dr + di * di + 1e-30f);
        for (int r = k + 1; r < 8; ++r) {
            float fr = (ar[r][k] * dr + ai[r][k] * di) * inv;
            float fi = (ai[r][k] * dr - ar[r][k] * di) * inv;
            for (int c = k; c < 9; ++c) {
                float br = ar[k][c], bi = ai[k][c];
                ar[r][c] -= fr * br - fi * bi;
                ai[r][c] -= fr * bi + fi * br;
            }
        }
    }
    float wr[8], wi[8];
    for (int k = 7; k >= 0; --k) {
        float sr = ar[k][8], si = ai[k][8];
        for (int c = k + 1; c < 8; ++c) {
            sr -= ar[k][c] * wr[c] - ai[k][c] * wi[c];
            si -= ar[k][c] * wi[c] + ai[k][c] * wr[c];
        }
        float dr = ar[k][k], di = ai[k][k];
        float inv = 1.0f / (dr * dr + di * di + 1e-30f);
        wr[k] = (sr * dr + si * di) * inv;
        wi[k] = (si * dr - sr * di) * inv;
    }
    float* o = wvec + (size_t)bin * 16;
    for (int m = 0; m < 8; ++m) { o[2 * m] = wr[m]; o[2 * m + 1] = wi[m]; }
}

// ---------------------------------------------------------------------------
// Kernel 3a: y = w^H x with the SAME TDM-staged tiling (coalesced x read),
// wave32 butterfly reduction over (re/im, mic), LDS transpose so y is written
// to workspace in coalesced rows, per-bin cov/var accumulated in registers.
// ---------------------------------------------------------------------------
__global__ __launch_bounds__(THREADS_COV)
void beamform_y_kernel(const float* __restrict__ xr, const float* __restrict__ xi,
                       const float* __restrict__ smask, const float* __restrict__ wvec,
                       float* __restrict__ y_ws, float* __restrict__ covvar)
{
    __shared__ float lx[2][2 * TILE_DW];
    __shared__ float lsm[CF][BPB];
    __shared__ float yt[CF][BPB][2];

    const int tid  = threadIdx.x;
    const int wave = tid >> 5;
    const int lane = tid & 31;
    const int half = lane >> 4;
    const int idx  = lane & 15;
    const int mic  = idx & 7;
    const int reim = idx >> 3;
    const int bin0 = blockIdx.x * BPB;
    const int bin  = bin0 + wave;
    const bool issuer = (wave == 0);

    float wr[8], wi[8];
    {
        const float* wv = wvec + (size_t)((bin < NBINS) ? bin : 0) * 16;
        #pragma unroll
        for (int m = 0; m < 8; ++m) { wr[m] = wv[2 * m]; wi[m] = wv[2 * m + 1]; }
    }

    float cr = 0.0f, ci = 0.0f, vv = 0.0f;

#if HAVE_TDM
    if (issuer) {
        tdm_load_tile(xr, 0, bin0, lds_off(&lx[0][0]));
        tdm_load_tile(xi, 0, bin0, lds_off(&lx[0][TILE_DW]));
    }
#endif

    int buf = 0;
    for (int f0 = 0; f0 < FRAMES; f0 += CF, buf ^= 1) {
        for (int e = tid; e < CF * BPB; e += THREADS_COV) {
            int bl = e & 15, f = e >> 4;
            int b = bin0 + bl;
            lsm[f][bl] = (b < NBINS) ? sigmoidf(smask[(size_t)(f0 + f) * NBINS + b]) : 0.0f;
        }
#if HAVE_TDM
        if (issuer) {
            if (f0 + CF < FRAMES) {
                tdm_load_tile(xr, f0 + CF, bin0, lds_off(&lx[buf ^ 1][0]));
                tdm_load_tile(xi, f0 + CF, bin0, lds_off(&lx[buf ^ 1][TILE_DW]));
                __builtin_amdgcn_s_wait_tensorcnt((short)2);
            } else {
                __builtin_amdgcn_s_wait_tensorcnt((short)0);
            }
        }
#else
        stage_x_fallback(&lx[buf][0], xr, xi, f0, bin0, tid);
#endif
        __syncthreads();

        const float* xs = &lx[buf][reim * TILE_DW];
        for (int s = 0; s < CF; s += 2) {
            int f = s + half;                       // each 16-lane half owns one frame
            float v = xs[f * FROW + mic * MROW + wave];
            float vim0 = __shfl_xor(v, 8, 32);      // on idx==0 lanes: x_im[f][mic0]
            // y = sum_m conj(w_m) x_m : ar = sum(wr*xr + wi*xi), ai = sum(wr*xi - wi*xr)
            float ca = reim ? wi[mic] * v :  wr[mic] * v;
            float cb = reim ? wr[mic] * v : -wi[mic] * v;
            #pragma unroll
            for (int off = 1; off < 16; off <<= 1) {
                ca += __shfl_xor(ca, off, 32);
                cb += __shfl_xor(cb, off, 32);
            }
            if (idx == 0) {                          // v = x_re[f][mic0] here
                float sm = lsm[f][wave];
                float tr = sm * v;
                float ti = sm * vim0;
                cr += tr * ca + ti * cb;             // cov += target * conj(y)
                ci += ti * ca - tr * cb;
                vv += ca * ca + cb * cb;             // var += |y|^2
                yt[f][wave][0] = ca;
                yt[f][wave][1] = cb;
            }
        }
        __syncthreads();
        // coalesced y rows: 16 bins * 8B contiguous per frame
        for (int e = tid; e < CF * BPB * 2; e += THREADS_COV) {
            int c = e & 1, bl = (e >> 1) & 15, f = e >> 5;
            int b = bin0 + bl;
            if (b < NBINS)
                y_ws[((size_t)(f0 + f) * NBINS + b) * 2 + c] = yt[f][bl][c];
        }
        __syncthreads();
    }

    #pragma unroll
    for (int off = 1; off < 32; off <<= 1) {
        cr += __shfl_xor(cr, off, 32);
        ci += __shfl_xor(ci, off, 32);
        vv += __shfl_xor(vv, off, 32);
    }
    if (lane == 0 && bin < NBINS) {
        covvar[bin * 4 + 0] = cr;
        covvar[bin * 4 + 1] = ci;
        covvar[bin * 4 + 2] = vv;
    }
}

// ---------------------------------------------------------------------------
// Kernel 3b: y *= cov/var, fully coalesced read of y_ws and write of out.
// ---------------------------------------------------------------------------
__global__ __launch_bounds__(256)
void scale_kernel(const float* __restrict__ y_ws, const float* __restrict__ covvar,
                  float* __restrict__ out)
{
    int f = blockIdx.y;
    int b = blockIdx.x * blockDim.x + threadIdx.x;
    if (b >= NBINS) return;
    float cr = covvar[b * 4], ci = covvar[b * 4 + 1], vv = covvar[b * 4 + 2];
    float iv = 1.0f / (vv + 1e-30f);
    float sr = cr * iv, si = ci * iv;
    size_t i = ((size_t)f * NBINS + b) * 2;
    float ar = y_ws[i], ai = y_ws[i + 1];
    out[i]     = ar * sr - ai * si;
    out[i + 1] = ar * si + ai * sr;
}

// ---------------------------------------------------------------------------
extern "C" void kernel_launch(void* const* d_in, const int* in_sizes, int n_in,
                              void* d_out, int out_size, void* d_ws, size_t ws_size,
                              hipStream_t stream)
{
    const float* xr    = (const float*)d_in[0];
    const float* xi    = (const float*)d_in[1];
    // d_in[2], d_in[3] (s_real, s_imag) are unused by the reference
    const float* masks = (const float*)d_in[4];
    const float* smask = (const float*)d_in[5];
    float* out = (float*)d_out;

    float* tiles  = (float*)d_ws;                       // NBINS*2*256 floats
    float* wvec   = tiles  + (size_t)NBINS * 2 * 256;   // NBINS*16
    float* covvar = wvec   + (size_t)NBINS * 16;        // NBINS*4
    float* y_ws   = covvar + (size_t)NBINS * 4;         // FRAMES*NBINS*2 (~16.8 MB)

    dim3 gtile((NBINS + BPB - 1) / BPB);
    cov_wmma_kernel<<<gtile, THREADS_COV, 0, stream>>>(xr, xi, masks, tiles);
    solve_kernel<<<(NBINS + 63) / 64, 64, 0, stream>>>(tiles, wvec);
    beamform_y_kernel<<<gtile, THREADS_COV, 0, stream>>>(xr, xi, smask, wvec, y_ws, covvar);
    dim3 gs((NBINS + 255) / 256, FRAMES);
    scale_kernel<<<gs, 256, 0, stream>>>(y_ws, covvar, out);
}